// Critic_6786048328272
// MI455X (gfx1250) — compile-verified
//
#include <hip/hip_runtime.h>
#include <hip/hip_bf16.h>

typedef __attribute__((ext_vector_type(2))) float v2f;
typedef __attribute__((ext_vector_type(8))) float v8f;

#define LN_EPS 1e-5f

// GEMM tiling
#define TM   128   // rows per block (8 waves x 16)
#define TN   64    // cols per block
#define KBLK 32    // K chunk
#define ASTR 36    // padded LDS stride for A rows (bank de-conflict, mult of 4)
#define BSTR 68    // padded LDS stride for B rows (bank de-conflict, mult of 4)

// ---------------------------------------------------------------------------
// Graph preprocessing
// ---------------------------------------------------------------------------
__global__ void k_count_nodes(const int* __restrict__ batch, int* __restrict__ cnt, int N) {
    int i = blockIdx.x * blockDim.x + threadIdx.x;
    if (i < N) atomicAdd(&cnt[batch[i]], 1);
}

__global__ void k_deg(const int* __restrict__ dst, float* __restrict__ deg, int E) {
    int e = blockIdx.x * blockDim.x + threadIdx.x;
    if (e < E) unsafeAtomicAdd(&deg[dst[e]], 1.0f);
}

__global__ void k_finalize_deg(const float* __restrict__ deg, float* __restrict__ dinv,
                               float* __restrict__ selfw, int N) {
    int i = blockIdx.x * blockDim.x + threadIdx.x;
    if (i < N) {
        float d = deg[i] + 1.0f;
        dinv[i]  = rsqrtf(d);
        selfw[i] = 1.0f / d;
    }
}

__global__ void k_edgew(const int* __restrict__ src, const int* __restrict__ dst,
                        const float* __restrict__ dinv, float* __restrict__ ew, int E) {
    int e = blockIdx.x * blockDim.x + threadIdx.x;
    if (e < E) ew[e] = dinv[src[e]] * dinv[dst[e]];
}

// last_idx[g] = searchsorted(batch, g, 'right') - 1 == cumsum(count)[g] - 1
__global__ void k_lastidx(const int* __restrict__ cnt, int* __restrict__ last, int B) {
    if (blockIdx.x == 0 && threadIdx.x == 0) {
        int c = 0;
        for (int g = 0; g < B; ++g) { c += cnt[g]; last[g] = c - 1; }
    }
}

// ---------------------------------------------------------------------------
// Dense GEMM, CDNA5 fp32 WMMA, double-buffered LDS staging via
// global_load_async_to_lds_b128 (ASYNCcnt) + ds_load fragment feeds.
// C[N,Hc] = A[N,K] @ Bm[K,Hc]   row-major; N%16==0, K%KBLK==0, Hc%TN==0.
// ---------------------------------------------------------------------------
__global__ __launch_bounds__(256)
void k_gemm_wmma(const float* __restrict__ A, const float* __restrict__ Bm,
                 float* __restrict__ C, int N, int K, int Hc) {
    __shared__ float As[2][TM * ASTR];   // 2 x 18432 B
    __shared__ float Bs[2][KBLK * BSTR]; // 2 x  8704 B   (total ~53 KB)

    const int t      = threadIdx.x;
    const int waveId = t >> 5;
    const int lane   = t & 31;
    const int half   = lane >> 4;   // 0 -> K pair {0,1}, 1 -> {2,3}
    const int lm     = lane & 15;
    const int h2     = half * 2;

    const int rowBase = blockIdx.x * TM;
    const int col0    = blockIdx.y * TN;
    const int row0    = rowBase + waveId * 16;
    const int nchunks = K / KBLK;

    // Issue one chunk's async DMA into LDS buffer (6 x b128 per thread).
    auto issue_chunk = [&](int c) {
        const int kc  = c * KBLK;
        const int buf = c & 1;
        unsigned abase = (unsigned)(size_t)&As[buf][0];
        unsigned bbase = (unsigned)(size_t)&Bs[buf][0];
        #pragma unroll
        for (int i = 0; i < 4; ++i) {               // A: 128x32 floats
            int lin = i * 256 + t;
            int r   = lin >> 3;
            int c4  = (lin & 7) * 4;
            int gr  = rowBase + r; if (gr > N - 1) gr = N - 1;  // clamp, keep EXEC uniform
            const float* gp = A + (size_t)gr * K + kc + c4;
            unsigned lo = abase + (unsigned)(r * ASTR + c4) * 4u;
            asm volatile("global_load_async_to_lds_b128 %0, %1, off"
                         :: "v"(lo), "v"((unsigned long long)(size_t)gp) : "memory");
        }
        #pragma unroll
        for (int i = 0; i < 2; ++i) {               // B: 32x64 floats
            int lin = i * 256 + t;
            int r   = lin >> 4;
            int c4  = (lin & 15) * 4;
            const float* gp = Bm + (size_t)(kc + r) * Hc + col0 + c4;
            unsigned lo = bbase + (unsigned)(r * BSTR + c4) * 4u;
            asm volatile("global_load_async_to_lds_b128 %0, %1, off"
                         :: "v"(lo), "v"((unsigned long long)(size_t)gp) : "memory");
        }
    };

    v8f acc0 = {0,0,0,0,0,0,0,0};
    v8f acc1 = {0,0,0,0,0,0,0,0};
    v8f acc2 = {0,0,0,0,0,0,0,0};
    v8f acc3 = {0,0,0,0,0,0,0,0};

    issue_chunk(0);

    for (int c = 0; c < nchunks; ++c) {
        if (c + 1 < nchunks) {
            issue_chunk(c + 1);
            // chunk c complete when only chunk c+1's 6 loads remain (in-order)
            asm volatile("s_wait_asynccnt 0x6" ::: "memory");
        } else {
            asm volatile("s_wait_asynccnt 0x0" ::: "memory");
        }
        __syncthreads();

        const int buf = c & 1;
        const float* __restrict__ as = &As[buf][(waveId * 16 + lm) * ASTR];
        const float* __restrict__ bs = &Bs[buf][0];

        #pragma unroll
        for (int kk = 0; kk < KBLK; kk += 4) {
            v2f a;
            a.x = as[kk + h2];
            a.y = as[kk + h2 + 1];

            const float* b0 = bs + (kk + h2) * BSTR + lm;
            const float* b1 = b0 + BSTR;

            v2f b;
            b.x = b0[0];  b.y = b1[0];
            acc0 = __builtin_amdgcn_wmma_f32_16x16x4_f32(false, a, false, b, (short)0, acc0, false, false);
            b.x = b0[16]; b.y = b1[16];
            acc1 = __builtin_amdgcn_wmma_f32_16x16x4_f32(false, a, false, b, (short)0, acc1, false, false);
            b.x = b0[32]; b.y = b1[32];
            acc2 = __builtin_amdgcn_wmma_f32_16x16x4_f32(false, a, false, b, (short)0, acc2, false, false);
            b.x = b0[48]; b.y = b1[48];
            acc3 = __builtin_amdgcn_wmma_f32_16x16x4_f32(false, a, false, b, (short)0, acc3, false, false);
        }
        __syncthreads();
    }

    // D layout: VGPR j -> (M = j + half*8, N = lm)
    if (row0 < N) {
        #pragma unroll
        for (int j = 0; j < 8; ++j) {
            int r = row0 + j + half * 8;
            float* cp = C + (size_t)r * Hc + col0 + lm;
            cp[0]  = acc0[j];
            cp[16] = acc1[j];
            cp[32] = acc2[j];
            cp[48] = acc3[j];
        }
    }
}

// ---------------------------------------------------------------------------
// SpMM: agg[dst] += h[src] * ew   (one wave32 per edge; H == 256, 8 f32/lane)
// ---------------------------------------------------------------------------
__global__ void k_spmm(const float* __restrict__ h, const int* __restrict__ src,
                       const int* __restrict__ dst, const float* __restrict__ ew,
                       float* __restrict__ agg, int E, int H) {
    const int wpb  = blockDim.x >> 5;
    const int wave = blockIdx.x * wpb + (threadIdx.x >> 5);
    const int tot  = gridDim.x * wpb;
    const int lane = threadIdx.x & 31;

    for (int e = wave; e < E; e += tot) {
        int nxt = e + tot;
        if (nxt < E) {
            // warm caches for next gather row -> global_prefetch_b8
            __builtin_prefetch(h + (size_t)src[nxt] * H + lane * 8, 0, 0);
        }
        const int   s = src[e];
        const int   d = dst[e];
        const float w = ew[e];
        const float4* hs = (const float4*)(h + (size_t)s * H);
        float* ad = agg + (size_t)d * H + lane * 8;
        float4 v0 = hs[lane * 2 + 0];
        float4 v1 = hs[lane * 2 + 1];
        unsafeAtomicAdd(ad + 0, v0.x * w);
        unsafeAtomicAdd(ad + 1, v0.y * w);
        unsafeAtomicAdd(ad + 2, v0.z * w);
        unsafeAtomicAdd(ad + 3, v0.w * w);
        unsafeAtomicAdd(ad + 4, v1.x * w);
        unsafeAtomicAdd(ad + 5, v1.y * w);
        unsafeAtomicAdd(ad + 6, v1.z * w);
        unsafeAtomicAdd(ad + 7, v1.w * w);
    }
}

// agg[i,:] += h[i,:] * selfw[i] + bias[:]   (one block per node)
__global__ void k_selfbias(float* __restrict__ agg, const float* __restrict__ h,
                           const float* __restrict__ selfw, const float* __restrict__ bias,
                           int H) {
    const int i = blockIdx.x;
    const int c = threadIdx.x;
    size_t idx = (size_t)i * H + c;
    agg[idx] += h[idx] * selfw[i] + bias[c];
}

// ---------------------------------------------------------------------------
// Graph LayerNorm: per-graph sum/sumsq over all nodes & channels
// ---------------------------------------------------------------------------
__global__ void k_stats(const float* __restrict__ x, const int* __restrict__ batch,
                        float* __restrict__ gsum, float* __restrict__ gsumsq,
                        int N, int H) {
    const int wpb  = blockDim.x >> 5;
    const int node = blockIdx.x * wpb + (threadIdx.x >> 5);
    if (node >= N) return;
    const int lane = threadIdx.x & 31;
    const float4* xr = (const float4*)(x + (size_t)node * H);
    float4 a = xr[lane * 2 + 0];
    float4 b = xr[lane * 2 + 1];
    float s  = a.x + a.y + a.z + a.w + b.x + b.y + b.z + b.w;
    float ss = a.x*a.x + a.y*a.y + a.z*a.z + a.w*a.w
             + b.x*b.x + b.y*b.y + b.z*b.z + b.w*b.w;
    #pragma unroll
    for (int off = 16; off > 0; off >>= 1) {
        s  += __shfl_down(s,  off, 32);
        ss += __shfl_down(ss, off, 32);
    }
    if (lane == 0) {
        int g = batch[node];
        unsafeAtomicAdd(&gsum[g],   s);
        unsafeAtomicAdd(&gsumsq[g], ss);
    }
}

__global__ void k_statfin(const float* __restrict__ gsum, const float* __restrict__ gsumsq,
                          const int* __restrict__ cnt, float* __restrict__ gmean,
                          float* __restrict__ ginv, int B, int H) {
    int g = blockIdx.x * blockDim.x + threadIdx.x;
    if (g < B) {
        float c = fmaxf((float)cnt[g], 1.0f);
        float denom = c * (float)H;
        float m   = gsum[g] / denom;
        float var = gsumsq[g] / denom - m * m;
        gmean[g] = m;
        ginv[g]  = rsqrtf(var + LN_EPS);
    }
}

// out = tanh((x - mean[g]) * inv[g] * w + b)   (one block per node)
__global__ void k_lnact(const float* __restrict__ x, const int* __restrict__ batch,
                        const float* __restrict__ gmean, const float* __restrict__ ginv,
                        const float* __restrict__ lw, const float* __restrict__ lb,
                        float* __restrict__ out, int H) {
    const int i = blockIdx.x;
    const int c = threadIdx.x;
    const int g = batch[i];
    size_t idx = (size_t)i * H + c;
    float v = (x[idx] - gmean[g]) * ginv[g] * lw[c] + lb[c];
    out[idx] = tanhf(v);
}

// xs_small[g, colOff + c] = ln[last[g], c]
__global__ void k_gather(const float* __restrict__ ln, const int* __restrict__ last,
                         float* __restrict__ xs, int colOff, int H, int HCAT) {
    const int g = blockIdx.x;
    const int c = threadIdx.x;
    xs[(size_t)g * HCAT + colOff + c] = ln[(size_t)last[g] * H + c];
}

// ---------------------------------------------------------------------------
// Head (only the B selected nodes): JK linear -> fc1+tanh -> fc2
// ---------------------------------------------------------------------------
__global__ __launch_bounds__(256)
void k_head(const float* __restrict__ xs, const float* __restrict__ jkw,
            const float* __restrict__ jkb, const float* __restrict__ f1w,
            const float* __restrict__ f1b, const float* __restrict__ f2w,
            const float* __restrict__ f2b, float* __restrict__ out,
            int H, int HCAT, int F1) {
    __shared__ float xrow[768];
    __shared__ float jk[256];
    __shared__ float f1[128];
    __shared__ float red[4];

    const int g = blockIdx.x;
    const int t = threadIdx.x;

    for (int i = t; i < HCAT; i += blockDim.x) xrow[i] = xs[(size_t)g * HCAT + i];
    __syncthreads();

    float acc = jkb[t];
    for (int k = 0; k < HCAT; ++k) acc += xrow[k] * jkw[(size_t)k * H + t];
    jk[t] = acc;
    __syncthreads();

    if (t < F1) {
        float a = f1b[t];
        for (int k = 0; k < H; ++k) a += jk[k] * f1w[(size_t)k * F1 + t];
        f1[t] = tanhf(a);
    }
    __syncthreads();

    if (t < F1) {
        float p = f1[t] * f2w[t];
        #pragma unroll
        for (int off = 16; off > 0; off >>= 1) p += __shfl_down(p, off, 32);
        if ((t & 31) == 0) red[t >> 5] = p;
    }
    __syncthreads();
    if (t == 0) out[g] = red[0] + red[1] + red[2] + red[3] + f2b[0];
}

// ---------------------------------------------------------------------------
// Host orchestration
// ---------------------------------------------------------------------------
static inline size_t align_up(size_t x, size_t a) { return (x + a - 1) & ~(a - 1); }

extern "C" void kernel_launch(void* const* d_in, const int* in_sizes, int n_in,
                              void* d_out, int out_size, void* d_ws, size_t ws_size,
                              hipStream_t stream) {
    const float* x     = (const float*)d_in[0];
    const float* W1    = (const float*)d_in[1];
    const float* b1    = (const float*)d_in[2];
    const float* ln1w  = (const float*)d_in[3];
    const float* ln1b  = (const float*)d_in[4];
    const float* W2    = (const float*)d_in[5];
    const float* b2    = (const float*)d_in[6];
    const float* ln2w  = (const float*)d_in[7];
    const float* ln2b  = (const float*)d_in[8];
    const float* W3    = (const float*)d_in[9];
    const float* b3    = (const float*)d_in[10];
    const float* ln3w  = (const float*)d_in[11];
    const float* ln3b  = (const float*)d_in[12];
    const float* jkw   = (const float*)d_in[13];
    const float* jkb   = (const float*)d_in[14];
    const float* fc1w  = (const float*)d_in[15];
    const float* fc1b  = (const float*)d_in[16];
    const float* fc2w  = (const float*)d_in[17];
    const float* fc2b  = (const float*)d_in[18];
    const int*   eidx  = (const int*)d_in[19];
    const int*   batch = (const int*)d_in[20];
    float*       outp  = (float*)d_out;

    const int H = in_sizes[2];          // 256
    const int D = in_sizes[1] / H;      // 128
    const int N = in_sizes[0] / D;      // 100000
    const int E = in_sizes[19] / 2;     // 1600000
    const int B = out_size;             // 64
    const int HCAT = 3 * H;
    const int F1 = H / 2;

    const int* src = eidx;
    const int* dst = eidx + E;

    // ---- carve workspace ----
    char* ws = (char*)d_ws;
    size_t off = 0;
    const size_t NH = (size_t)N * H * sizeof(float);
    float* U      = (float*)(ws + off); off = align_up(off + NH, 256);
    float* V      = (float*)(ws + off); off = align_up(off + NH, 256);
    float* deg    = (float*)(ws + off); off = align_up(off + (size_t)N * 4, 256);
    float* dinv   = (float*)(ws + off); off = align_up(off + (size_t)N * 4, 256);
    float* selfw  = (float*)(ws + off); off = align_up(off + (size_t)N * 4, 256);
    float* ew     = (float*)(ws + off); off = align_up(off + (size_t)E * 4, 256);
    int*   gcnt   = (int*)  (ws + off); off = align_up(off + (size_t)B * 4, 256);
    float* gsum   = (float*)(ws + off);   // gsum & gsumsq contiguous (one memset)
    float* gsumsq = gsum + B;           off = align_up(off + (size_t)2 * B * 4, 256);
    float* gmean  = (float*)(ws + off); off = align_up(off + (size_t)B * 4, 256);
    float* ginv   = (float*)(ws + off); off = align_up(off + (size_t)B * 4, 256);
    int*   lastix = (int*)  (ws + off); off = align_up(off + (size_t)B * 4, 256);
    float* xs_sm  = (float*)(ws + off); off = align_up(off + (size_t)B * HCAT * 4, 256);
    (void)ws_size;

    // ---- preprocessing ----
    hipMemsetAsync(deg,  0, (size_t)N * 4, stream);
    hipMemsetAsync(gcnt, 0, (size_t)B * 4, stream);

    k_count_nodes<<<(N + 255) / 256, 256, 0, stream>>>(batch, gcnt, N);
    k_deg<<<(E + 255) / 256, 256, 0, stream>>>(dst, deg, E);
    k_finalize_deg<<<(N + 255) / 256, 256, 0, stream>>>(deg, dinv, selfw, N);
    k_edgew<<<(E + 255) / 256, 256, 0, stream>>>(src, dst, dinv, ew, E);
    k_lastidx<<<1, 1, 0, stream>>>(gcnt, lastix, B);

    const dim3 gemmBlk(256);
    const int  spmmBlocks  = ((E + 7) / 8 < 20000) ? (E + 7) / 8 : 20000;
    const int  statsBlocks = (N + 7) / 8;

    // ===== Layer 1: in = x[N,D], gemm->U, agg->V, ln->U =====
    {
        dim3 g1((N + TM - 1) / TM, H / TN);
        k_gemm_wmma<<<g1, gemmBlk, 0, stream>>>(x, W1, U, N, D, H);
        hipMemsetAsync(V, 0, NH, stream);
        k_spmm<<<spmmBlocks, 256, 0, stream>>>(U, src, dst, ew, V, E, H);
        k_selfbias<<<N, H, 0, stream>>>(V, U, selfw, b1, H);
        hipMemsetAsync(gsum, 0, (size_t)2 * B * 4, stream);
        k_stats<<<statsBlocks, 256, 0, stream>>>(V, batch, gsum, gsumsq, N, H);
        k_statfin<<<1, 64, 0, stream>>>(gsum, gsumsq, gcnt, gmean, ginv, B, H);
        k_lnact<<<N, H, 0, stream>>>(V, batch, gmean, ginv, ln1w, ln1b, U, H);
        k_gather<<<B, H, 0, stream>>>(U, lastix, xs_sm, 0, H, HCAT);
    }

    // ===== Layer 2: in = U, gemm->V, agg->U, ln->V =====
    {
        dim3 g2((N + TM - 1) / TM, H / TN);
        k_gemm_wmma<<<g2, gemmBlk, 0, stream>>>(U, W2, V, N, H, H);
        hipMemsetAsync(U, 0, NH, stream);
        k_spmm<<<spmmBlocks, 256, 0, stream>>>(V, src, dst, ew, U, E, H);
        k_selfbias<<<N, H, 0, stream>>>(U, V, selfw, b2, H);
        hipMemsetAsync(gsum, 0, (size_t)2 * B * 4, stream);
        k_stats<<<statsBlocks, 256, 0, stream>>>(U, batch, gsum, gsumsq, N, H);
        k_statfin<<<1, 64, 0, stream>>>(gsum, gsumsq, gcnt, gmean, ginv, B, H);
        k_lnact<<<N, H, 0, stream>>>(U, batch, gmean, ginv, ln2w, ln2b, V, H);
        k_gather<<<B, H, 0, stream>>>(V, lastix, xs_sm, H, H, HCAT);
    }

    // ===== Layer 3: in = V, gemm->U, agg->V, ln->U =====
    {
        dim3 g3((N + TM - 1) / TM, H / TN);
        k_gemm_wmma<<<g3, gemmBlk, 0, stream>>>(V, W3, U, N, H, H);
        hipMemsetAsync(V, 0, NH, stream);
        k_spmm<<<spmmBlocks, 256, 0, stream>>>(U, src, dst, ew, V, E, H);
        k_selfbias<<<N, H, 0, stream>>>(V, U, selfw, b3, H);
        hipMemsetAsync(gsum, 0, (size_t)2 * B * 4, stream);
        k_stats<<<statsBlocks, 256, 0, stream>>>(V, batch, gsum, gsumsq, N, H);
        k_statfin<<<1, 64, 0, stream>>>(gsum, gsumsq, gcnt, gmean, ginv, B, H);
        k_lnact<<<N, H, 0, stream>>>(V, batch, gmean, ginv, ln3w, ln3b, U, H);
        k_gather<<<B, H, 0, stream>>>(U, lastix, xs_sm, 2 * H, H, HCAT);
    }

    // ===== Head on the B selected nodes only =====
    k_head<<<B, 256, 0, stream>>>(xs_sm, jkw, jkb, fc1w, fc1b, fc2w, fc2b, outp,
                                  H, HCAT, F1);
}